// GaussianSplattingModel_55774445306111
// MI455X (gfx1250) — compile-verified
//
#include <hip/hip_runtime.h>
#include <stdint.h>

typedef float v2f __attribute__((ext_vector_type(2)));
typedef float v8f __attribute__((ext_vector_type(8)));

#define N_CLUSTERS   8192
#define CLUSTER_SIZE 128
#define PLANE        (N_CLUSTERS * CLUSTER_SIZE)   /* 1,048,576 */

// ---------------------------------------------------------------------------
// Kernel A: frustum culling as a WMMA GEMM.
//   dist+rad (16 clusters x 6 planes) = [origin,1](16x4) x [n;d](4x16)
//                                     + [extend,0](16x4) x [|n|;0](4x16)
// A layout (f32 16x4, ISA 7.12.2): lanes 0-15 rows M, VGPR0 = K0(lo)/K2(hi),
// VGPR1 = K1(lo)/K3(hi). B (4x16) mirrored. C/D: VGPR v -> rows v (lanes0-15)
// and v+8 (lanes16-31), N = lane&15.
// ---------------------------------------------------------------------------
__global__ __launch_bounds__(32)
void cull_wmma_kernel(const float* __restrict__ org, const float* __restrict__ ext,
                      const float* __restrict__ fr, int* __restrict__ flags)
{
  const int  lane = threadIdx.x;          // one full wave32 => EXEC all ones
  const bool hi   = lane >= 16;
  const int  pl   = lane & 15;            // plane column N
  const int  plc  = (pl < 6) ? pl : 0;    // clamp to avoid OOB read
  float nx = fr[plc * 4 + 0];
  float ny = fr[plc * 4 + 1];
  float nz = fr[plc * 4 + 2];
  float dd = fr[plc * 4 + 3];
  if (pl >= 6) { nx = 0.f; ny = 0.f; nz = 0.f; dd = 0.f; }   // zero-pad N>=6

  v2f bn, ba;
  bn.x = hi ? nz : nx;                    // K2 : K0
  bn.y = hi ? dd : ny;                    // K3 : K1   (d row pairs with A's 1.0)
  ba.x = hi ? __builtin_fabsf(nz) : __builtin_fabsf(nx);
  ba.y = hi ? 0.0f : __builtin_fabsf(ny);

  for (int t = 0; t < 32; ++t) {
    const int m0  = (blockIdx.x * 32 + t) * 16;
    const int row = m0 + (lane & 15);
    const float ox = org[row * 3 + 0], oy = org[row * 3 + 1], oz = org[row * 3 + 2];
    const float ex = ext[row * 3 + 0], ey = ext[row * 3 + 1], ez = ext[row * 3 + 2];
    v2f ao, ae;
    ao.x = hi ? oz : ox;  ao.y = hi ? 1.0f : oy;     // [x y z 1]
    ae.x = hi ? ez : ex;  ae.y = hi ? 0.0f : ey;     // [ex ey ez 0]

    v8f acc = {};
    acc = __builtin_amdgcn_wmma_f32_16x16x4_f32(false, ae, false, ba,
                                                (short)0, acc, false, false);
    acc = __builtin_amdgcn_wmma_f32_16x16x4_f32(false, ao, false, bn,
                                                (short)0, acc, false, false);

    int myflag = 0;
#pragma unroll
    for (int v = 0; v < 8; ++v) {
      const bool     pred = acc[v] > 0.0f;
      const unsigned m    = (unsigned)__ballot(pred);
      const int okLo = ((m         & 0x3Fu) == 0x3Fu);  // row v   : planes 0..5
      const int okHi = (((m >> 16) & 0x3Fu) == 0x3Fu);  // row v+8 : planes 0..5
      if (lane == v)     myflag = okLo;
      if (lane == v + 8) myflag = okHi;
    }
    if (lane < 16) flags[m0 + lane] = myflag;
  }
}

// ---------------------------------------------------------------------------
// Kernel B: stable partition (visible clusters first) with one wave32,
// two-pass ballot/popcount scan over 8192 flags.
// ---------------------------------------------------------------------------
__global__ __launch_bounds__(32)
void scan_kernel(const int* __restrict__ flags, const int* __restrict__ idx_tensor,
                 int* __restrict__ order, int* __restrict__ numvis,
                 int* __restrict__ out_chunkid, int* __restrict__ out_nums)
{
  __shared__ unsigned ball[N_CLUSTERS / 32];
  const int lane = threadIdx.x;

  unsigned total = 0;
  for (int c = 0; c < N_CLUSTERS / 32; ++c) {
    const int      vis = flags[c * 32 + lane];
    const unsigned m   = (unsigned)__ballot(vis != 0);
    if (lane == 0) ball[c] = m;
    total += __popc(m);
  }

  const unsigned lt = (lane == 0) ? 0u : (0xFFFFFFFFu >> (32 - lane));
  unsigned runV = 0, runI = 0;
  for (int c = 0; c < N_CLUSTERS / 32; ++c) {
    const unsigned m   = ball[c];
    const int      k   = c * 32 + lane;
    const bool     vis = (m >> lane) & 1u;
    const unsigned pv  = __popc(m & lt);
    const unsigned pi  = (unsigned)lane - pv;
    const unsigned pos = vis ? (runV + pv) : (total + runI + pi);
    order[pos]       = k;
    out_chunkid[pos] = idx_tensor[k];
    const unsigned cnt = __popc(m);
    runV += cnt;
    runI += 32u - cnt;
  }
  if (lane == 0) {
    numvis[0]   = (int)total;
    out_nums[0] = (int)total;                      // visible_chunks_num
    out_nums[1] = (int)total * CLUSTER_SIZE;       // valid_length
  }
}

// ---------------------------------------------------------------------------
// Kernel C: the HBM-bound gather/transform. 2^20 threads, p (output position)
// fastest -> fully coalesced NT stores; gathered reads are near-sequential.
// order[] tile staged in LDS via CDNA5 async-to-LDS (ASYNCcnt).
// ---------------------------------------------------------------------------
__global__ __launch_bounds__(256)
void gather_kernel(const float* __restrict__ xyz,  const float* __restrict__ scl,
                   const float* __restrict__ rot,  const float* __restrict__ sh0,
                   const float* __restrict__ shr,  const float* __restrict__ opa,
                   const float* __restrict__ vm,
                   const int* __restrict__ order,  const int* __restrict__ numvis_p,
                   float* __restrict__ out_xyz,    float* __restrict__ out_scale,
                   float* __restrict__ out_rot,    float* __restrict__ out_color,
                   float* __restrict__ out_opac)
{
  __shared__ int s_order[256];
  const int tid = blockIdx.x * 256 + threadIdx.x;
  const int p   = tid & (N_CLUSTERS - 1);   // output cluster slot (block-contiguous)
  const int i   = tid >> 13;                // gaussian index within cluster

  // Async copy of this block's 256-entry order[] tile into LDS.
  {
    const unsigned lds = (unsigned)(size_t)(void*)&s_order[threadIdx.x];
    const int*     ga  = order + p;
    asm volatile("global_load_async_to_lds_b32 %0, %1, off"
                 :: "v"(lds), "v"(ga) : "memory");
  }

  const int   numvis = numvis_p[0];
  // campos = -(R^T t) from the 4x4 row-major view matrix (uniform, cached)
  const float camx = -(vm[0] * vm[3] + vm[4] * vm[7] + vm[8]  * vm[11]);
  const float camy = -(vm[1] * vm[3] + vm[5] * vm[7] + vm[9]  * vm[11]);
  const float camz = -(vm[2] * vm[3] + vm[6] * vm[7] + vm[10] * vm[11]);

  asm volatile("s_wait_asynccnt 0" ::: "memory");
  __syncthreads();

  const int    src   = s_order[threadIdx.x];
  const float  valid = (p < numvis) ? 1.0f : 0.0f;
  const size_t base  = (size_t)i * N_CLUSTERS + (size_t)src;
  const size_t obase = (size_t)i * N_CLUSTERS + (size_t)p;

  // Get SH cachelines in flight early (global_prefetch_b8).
  __builtin_prefetch(shr + base, 0, 0);
  __builtin_prefetch(shr + (size_t)24 * PLANE + base, 0, 0);

  // xyz ------------------------------------------------------------------
  const float gx = __builtin_nontemporal_load(xyz + base);
  const float gy = __builtin_nontemporal_load(xyz + (size_t)PLANE + base);
  const float gz = __builtin_nontemporal_load(xyz + (size_t)2 * PLANE + base);
  __builtin_nontemporal_store(gx * valid, out_xyz + obase);
  __builtin_nontemporal_store(gy * valid, out_xyz + (size_t)PLANE + obase);
  __builtin_nontemporal_store(gz * valid, out_xyz + (size_t)2 * PLANE + obase);

  // scale ----------------------------------------------------------------
#pragma unroll
  for (int c = 0; c < 3; ++c) {
    const float s = expf(__builtin_nontemporal_load(scl + (size_t)c * PLANE + base));
    __builtin_nontemporal_store(s * valid, out_scale + (size_t)c * PLANE + obase);
  }

  // rot (normalize) ------------------------------------------------------
  {
    const float r0 = __builtin_nontemporal_load(rot + base);
    const float r1 = __builtin_nontemporal_load(rot + (size_t)PLANE + base);
    const float r2 = __builtin_nontemporal_load(rot + (size_t)2 * PLANE + base);
    const float r3 = __builtin_nontemporal_load(rot + (size_t)3 * PLANE + base);
    const float inv = valid / sqrtf(r0 * r0 + r1 * r1 + r2 * r2 + r3 * r3);
    __builtin_nontemporal_store(r0 * inv, out_rot + obase);
    __builtin_nontemporal_store(r1 * inv, out_rot + (size_t)PLANE + obase);
    __builtin_nontemporal_store(r2 * inv, out_rot + (size_t)2 * PLANE + obase);
    __builtin_nontemporal_store(r3 * inv, out_rot + (size_t)3 * PLANE + obase);
  }

  // opacity (sigmoid) ----------------------------------------------------
  {
    const float o = __builtin_nontemporal_load(opa + base);
    __builtin_nontemporal_store(valid / (1.0f + expf(-o)), out_opac + obase);
  }

  // SH degree-3 color ----------------------------------------------------
  float dx = gx - camx, dy = gy - camy, dz = gz - camz;
  const float dinv = 1.0f / sqrtf(dx * dx + dy * dy + dz * dz);
  const float x = dx * dinv, y = dy * dinv, z = dz * dinv;
  const float xx = x * x, yy = y * y, zz = z * z;
  const float xy = x * y, yz = y * z, xz = x * z;

  float b[16];
  b[0]  = 0.28209479177387814f;
  b[1]  = -0.4886025119029199f * y;
  b[2]  =  0.4886025119029199f * z;
  b[3]  = -0.4886025119029199f * x;
  b[4]  =  1.0925484305920792f * xy;
  b[5]  = -1.0925484305920792f * yz;
  b[6]  =  0.31539156525252005f * (2.0f * zz - xx - yy);
  b[7]  = -1.0925484305920792f * xz;
  b[8]  =  0.5462742152960396f * (xx - yy);
  b[9]  = -0.5900435899266435f * y * (3.0f * xx - yy);
  b[10] =  2.890611442640554f  * xy * z;
  b[11] = -0.4570457994644658f * y * (4.0f * zz - xx - yy);
  b[12] =  0.3731763325901154f * z * (2.0f * zz - 3.0f * xx - 3.0f * yy);
  b[13] = -0.4570457994644658f * x * (4.0f * zz - xx - yy);
  b[14] =  1.445305721320277f  * z * (xx - yy);
  b[15] = -0.5900435899266435f * x * (xx - yy - zz);

#pragma unroll
  for (int c = 0; c < 3; ++c) {
    float acc = b[0] * __builtin_nontemporal_load(sh0 + (size_t)c * PLANE + base);
#pragma unroll
    for (int j = 1; j < 16; ++j) {
      const float coef =
          __builtin_nontemporal_load(shr + ((size_t)(j - 1) * 3 + c) * PLANE + base);
      acc = fmaf(b[j], coef, acc);
    }
    const float col = fmaxf(acc + 0.5f, 0.0f) * valid;
    __builtin_nontemporal_store(col, out_color + (size_t)c * PLANE + obase);
  }
}

// ---------------------------------------------------------------------------
extern "C" void kernel_launch(void* const* d_in, const int* in_sizes, int n_in,
                              void* d_out, int out_size, void* d_ws, size_t ws_size,
                              hipStream_t stream)
{
  const float* cluster_origin = (const float*)d_in[0];
  const float* cluster_extend = (const float*)d_in[1];
  const float* view_matrix    = (const float*)d_in[2];
  const float* frustumplane   = (const float*)d_in[3];
  const float* xyz            = (const float*)d_in[4];
  const float* scale          = (const float*)d_in[5];
  const float* rot            = (const float*)d_in[6];
  const float* sh_0           = (const float*)d_in[7];
  const float* sh_rest        = (const float*)d_in[8];
  const float* opacity        = (const float*)d_in[9];
  const int*   idx_tensor     = (const int*)d_in[10];
  (void)in_sizes; (void)n_in; (void)out_size; (void)ws_size;

  int* ws    = (int*)d_ws;
  int* flags = ws;                       // 8192 ints
  int* order = ws + N_CLUSTERS;          // 8192 ints
  int* numv  = ws + 2 * N_CLUSTERS;      // 1 int

  // Output layout (flat, reference return order):
  // [0,8192) visible_chunkid (i32 bits) | [8192,8194) counts (i32 bits) |
  // xyz 3P | scale 3P | rot 4P | color 3P | opacity P   (P = 2^20 floats)
  float* outf        = (float*)d_out;
  int*   out_chunkid = (int*)outf;
  int*   out_nums    = (int*)(outf + N_CLUSTERS);
  float* out_xyz     = outf + N_CLUSTERS + 2;
  float* out_scale   = out_xyz   + (size_t)3 * PLANE;
  float* out_rot     = out_scale + (size_t)3 * PLANE;
  float* out_color   = out_rot   + (size_t)4 * PLANE;
  float* out_opac    = out_color + (size_t)3 * PLANE;

  cull_wmma_kernel<<<16, 32, 0, stream>>>(cluster_origin, cluster_extend,
                                          frustumplane, flags);
  scan_kernel<<<1, 32, 0, stream>>>(flags, idx_tensor, order, numv,
                                    out_chunkid, out_nums);
  gather_kernel<<<PLANE / 256, 256, 0, stream>>>(xyz, scale, rot, sh_0, sh_rest,
                                                 opacity, view_matrix, order, numv,
                                                 out_xyz, out_scale, out_rot,
                                                 out_color, out_opac);
}